// GPStream_31001073942543
// MI455X (gfx1250) — compile-verified
//
#include <hip/hip_runtime.h>
#include <hip/hip_bf16.h>
#include <math.h>

typedef __attribute__((ext_vector_type(2))) float v2f;
typedef __attribute__((ext_vector_type(8))) float v8f;
typedef __attribute__((ext_vector_type(4))) unsigned int v4u;
typedef __attribute__((ext_vector_type(8))) int v8i;
typedef __attribute__((ext_vector_type(4))) int v4i;

#define M_HEADS 8
#define NCTX    1024
#define NTGT    512
#define DXF     3
#define NB      16
#define CHUNK0  512
#define NSTEPS  8
#define LRATE   0.1f
#define BETA1   0.9f
#define BETA2   0.999f
#define EPSADAM 1e-8f
#define SQRT5F  2.2360679774997896f
#define NOISELB 1e-4f

// ---- Tensor Data Mover availability (device pass only; host pass falls back)
#if defined(__HIP_DEVICE_COMPILE__) && __has_builtin(__builtin_amdgcn_tensor_load_to_lds)
#define HAVE_TDM 1
#else
#define HAVE_TDM 0
#endif

#if __has_include(<hip/amd_detail/amd_gfx1250_TDM.h>)
#define TDM_SIX_ARGS 1
#else
#define TDM_SIX_ARGS 0
#endif

#if HAVE_TDM
__device__ __forceinline__ unsigned lds_addr32(const void* p) {
  // LDS byte address for D#.lds_addr (flat-address low bits)
  return (unsigned)(unsigned long long)p;
}

// 2D tile DMA: rows x cols fp32 elements, row stride `stride` elements,
// packed contiguously into LDS at lds_byte.
__device__ __forceinline__ void tdm_load_2d(const void* gsrc, unsigned lds_byte,
                                            unsigned rows, unsigned cols,
                                            unsigned stride) {
  unsigned long long ga = (unsigned long long)gsrc;
  v4u g0;
  g0[0] = 1u;                                                  // count=1 (user D#)
  g0[1] = lds_byte;                                            // lds_addr
  g0[2] = (unsigned)ga;                                        // global_addr lo
  g0[3] = (unsigned)((ga >> 32) & 0x01ffffffu) | (2u << 30);   // addr hi | type=2
  v8i g1;
  g1[0] = (int)(2u << 16);                                     // data_size=4B
  g1[1] = (int)((stride & 0xffffu) << 16);                     // tensor_dim0[15:0]
  g1[2] = (int)(((stride >> 16) & 0xffffu) |                   // tensor_dim0[31:16]
                ((rows & 0xffffu) << 16));                     // tensor_dim1[15:0]
  g1[3] = (int)(((rows >> 16) & 0xffffu) |                     // tensor_dim1[31:16]
                ((cols & 0xffffu) << 16));                     // tile_dim0
  g1[4] = (int)(rows & 0xffffu);                               // tile_dim1 (tile_dim2=0)
  g1[5] = (int)stride;                                         // tensor_dim0_stride lo
  g1[6] = 0;
  g1[7] = 0;
  v4i z4 = {0, 0, 0, 0};
#if TDM_SIX_ARGS
  v8i z8 = {0, 0, 0, 0, 0, 0, 0, 0};
  __builtin_amdgcn_tensor_load_to_lds(g0, g1, z4, z4, z8, 0);
#else
  __builtin_amdgcn_tensor_load_to_lds(g0, g1, z4, z4, 0);
#endif
}
#endif  // HAVE_TDM

__device__ __forceinline__ float softplusf(float x) {
  return (x > 20.f) ? x : log1pf(expf(x));
}

__device__ __forceinline__ float block_reduce256(float v, float* red, int tid) {
  red[tid] = v;
  __syncthreads();
  for (int s = 128; s > 0; s >>= 1) {
    if (tid < s) red[tid] += red[tid + s];
    __syncthreads();
  }
  float r = red[0];
  __syncthreads();
  return r;
}

// ---------------------------------------------------------------- init params
__global__ void init_params(float* ws) {
  int i = threadIdx.x;
  if (i < M_HEADS * 4) {
    ws[i] = 0.f;        // params
    ws[32 + i] = 0.f;   // adam m
    ws[64 + i] = 0.f;   // adam v
  }
}

// ---------------------------------------------------------------- K(xc,xc)
__global__ __launch_bounds__(256) void build_K(const float* __restrict__ xc,
                                               float* __restrict__ Kbuf,
                                               const float* __restrict__ par,
                                               int N) {
  int m = blockIdx.z;
  int i = blockIdx.y * 16 + (threadIdx.x >> 4);
  int j = blockIdx.x * 16 + (threadIdx.x & 15);
  float ls = softplusf(par[m * 4 + 0]);
  float os = softplusf(par[m * 4 + 1]);
  float nz = softplusf(par[m * 4 + 2]) + NOISELB;
  const float* xi = xc + ((size_t)m * NCTX + i) * DXF;
  const float* xj = xc + ((size_t)m * NCTX + j) * DXF;
  float d2 = 0.f;
#pragma unroll
  for (int d = 0; d < DXF; ++d) {
    float t = (xi[d] - xj[d]) / ls;
    d2 += t * t;
  }
  d2 = fmaxf(d2, 1e-12f);
  float dd = sqrtf(d2);
  float k = os * (1.f + SQRT5F * dd + (5.f / 3.f) * d2) * expf(-SQRT5F * dd);
  if (i == j) k += nz;
  Kbuf[(size_t)m * NCTX * NCTX + (size_t)i * N + j] = k;
}

// ---------------------------------------------------------------- K(xc,xt) -> B (NCTX x NTGT)
__global__ __launch_bounds__(256) void build_Kct(const float* __restrict__ xc,
                                                 const float* __restrict__ xt,
                                                 float* __restrict__ B,
                                                 const float* __restrict__ par) {
  int m = blockIdx.z;
  int n = blockIdx.y * 16 + (threadIdx.x >> 4);
  int t = blockIdx.x * 16 + (threadIdx.x & 15);
  float ls = softplusf(par[m * 4 + 0]);
  float os = softplusf(par[m * 4 + 1]);
  const float* xi = xc + ((size_t)m * NCTX + n) * DXF;
  const float* xj = xt + ((size_t)m * NTGT + t) * DXF;
  float d2 = 0.f;
#pragma unroll
  for (int d = 0; d < DXF; ++d) {
    float s = (xi[d] - xj[d]) / ls;
    d2 += s * s;
  }
  d2 = fmaxf(d2, 1e-12f);
  float dd = sqrtf(d2);
  float k = os * (1.f + SQRT5F * dd + (5.f / 3.f) * d2) * expf(-SQRT5F * dd);
  B[(size_t)m * NCTX * NTGT + (size_t)n * NTGT + t] = k;
}

// ---------------------------------------------------------------- blocked Cholesky (in place)
// one workgroup (8 waves) per head; block-column staged to LDS via TDM;
// trailing SYRK via V_WMMA_F32_16X16X4_F32
__global__ __launch_bounds__(256) void chol_inplace(float* __restrict__ Kall, int N) {
  extern __shared__ float panel[];  // (N/16) tiles of 16x16 (block-column k)
  int m = blockIdx.x;
  float* K = Kall + (size_t)m * NCTX * NCTX;
  const int nb = N / NB;
  int tid = threadIdx.x;
  int lane = tid & 31, wave = tid >> 5;

  for (int k = 0; k < nb; ++k) {
    float* diag = panel + k * 256;
    // ---- stage raw block-column K[k*16.., k*16..k*16+16) into LDS
#if HAVE_TDM
    if (tid == 0) {
      tdm_load_2d(&K[(size_t)(k * 16) * N + k * 16], lds_addr32(panel + k * 256),
                  (unsigned)(N - k * 16), 16u, (unsigned)N);
      __builtin_amdgcn_s_wait_tensorcnt(0);
    }
#else
    for (int e = tid + k * 256; e < nb * 256; e += 256) {
      int t = e >> 8;
      panel[e] = K[(size_t)(t * 16 + ((e >> 4) & 15)) * N + k * 16 + (e & 15)];
    }
#endif
    __syncthreads();
    // ---- factor 16x16 diagonal (serial, tiny)
    if (tid == 0) {
      for (int j = 0; j < 16; ++j) {
        float d = sqrtf(fmaxf(diag[j * 16 + j], 1e-10f));
        diag[j * 16 + j] = d;
        float inv = 1.f / d;
        for (int i = j + 1; i < 16; ++i) diag[i * 16 + j] *= inv;
        for (int jj = j + 1; jj < 16; ++jj)
          for (int i = jj; i < 16; ++i)
            diag[i * 16 + jj] -= diag[i * 16 + j] * diag[jj * 16 + j];
      }
    }
    __syncthreads();
    // write factored diagonal back
    K[(size_t)(k * 16 + (tid >> 4)) * N + k * 16 + (tid & 15)] = diag[tid];
    // ---- panel TRSM: each thread owns one row of a sub-diagonal tile (LDS source)
    int r = tid & 15;
    for (int t = k + 1 + (tid >> 4); t < nb; t += 16) {
      float* pt = panel + t * 256;
      float4* prow = (float4*)&pt[r * 16];
      float row[16];
#pragma unroll
      for (int q = 0; q < 4; ++q) {
        float4 v = prow[q];
        row[q * 4 + 0] = v.x; row[q * 4 + 1] = v.y;
        row[q * 4 + 2] = v.z; row[q * 4 + 3] = v.w;
      }
#pragma unroll
      for (int c = 0; c < 16; ++c) {
        float s = row[c];
        for (int p = 0; p < c; ++p) s -= row[p] * diag[c * 16 + p];
        row[c] = s / diag[c * 16 + c];
      }
      float4* grow = (float4*)&K[(size_t)(t * 16 + r) * N + k * 16];
#pragma unroll
      for (int q = 0; q < 4; ++q) {
        float4 v = make_float4(row[q * 4 + 0], row[q * 4 + 1],
                               row[q * 4 + 2], row[q * 4 + 3]);
        prow[q] = v;
        grow[q] = v;
      }
    }
    __syncthreads();
    // ---- trailing SYRK: C_ij -= P_i * P_j^T  (i >= j > k), WMMA f32 16x16x4
    int rem = nb - (k + 1);
    int ntiles = rem * (rem + 1) / 2;
    for (int idx = wave; idx < ntiles; idx += 8) {
      int i_loc = (int)((sqrtf(8.0f * (float)idx + 1.0f) - 1.0f) * 0.5f);
      while ((i_loc + 1) * (i_loc + 2) / 2 <= idx) ++i_loc;
      while (i_loc * (i_loc + 1) / 2 > idx) --i_loc;
      int j_loc = idx - i_loc * (i_loc + 1) / 2;
      int i = k + 1 + i_loc, j = k + 1 + j_loc;
      const float* Pi = panel + i * 256;
      const float* Pj = panel + j * 256;
      int cl = lane & 15, ch = lane >> 4;
      float* Cp = &K[(size_t)(i * 16) * N + j * 16];
      __builtin_prefetch((const void*)Cp, 0, 1);
      v8f c;
#pragma unroll
      for (int v = 0; v < 8; ++v) c[v] = Cp[(size_t)(v + 8 * ch) * N + cl];
#pragma unroll
      for (int kk = 0; kk < 16; kk += 4) {
        int k0 = kk + (ch ? 2 : 0);
        v2f a, b;
        a.x = -Pi[cl * 16 + k0];       // A = -P_i  (16x4 slice)
        a.y = -Pi[cl * 16 + k0 + 1];
        b.x = Pj[cl * 16 + k0];        // B = P_j^T (4x16 slice)
        b.y = Pj[cl * 16 + k0 + 1];
        c = __builtin_amdgcn_wmma_f32_16x16x4_f32(false, a, false, b,
                                                  (short)0, c, false, false);
      }
#pragma unroll
      for (int v = 0; v < 8; ++v) Cp[(size_t)(v + 8 * ch) * N + cl] = c[v];
    }
    // make trailing-update stores visible before next iteration's TDM re-read
    asm volatile("s_wait_storecnt 0x0" ::: "memory");
    __syncthreads();
  }
}

// ---------------------------------------------------------------- cho_solve for alpha (1 rhs)
__global__ __launch_bounds__(256) void solve_alpha(const float* __restrict__ Kbuf,
                                                   const float* __restrict__ yc,
                                                   const float* __restrict__ par,
                                                   float* __restrict__ alpha, int N) {
  __shared__ float x[NCTX];
  int m = blockIdx.x;
  const float* L = Kbuf + (size_t)m * NCTX * NCTX;
  int tid = threadIdx.x;
  float cm = par[m * 4 + 3];
  for (int i = tid; i < N; i += 256) x[i] = yc[(size_t)m * NCTX + i] - cm;
  __syncthreads();
  for (int j = 0; j < N; ++j) {            // L x = b
    if (tid == 0) x[j] = x[j] / L[(size_t)j * N + j];
    __syncthreads();
    float xj = x[j];
    for (int i = j + 1 + tid; i < N; i += 256) x[i] -= L[(size_t)i * N + j] * xj;
    __syncthreads();
  }
  for (int j = N - 1; j >= 0; --j) {       // L^T a = x
    if (tid == 0) x[j] = x[j] / L[(size_t)j * N + j];
    __syncthreads();
    float xj = x[j];
    for (int i = tid; i < j; i += 256) x[i] -= L[(size_t)j * N + i] * xj;
    __syncthreads();
  }
  for (int i = tid; i < N; i += 256) alpha[(size_t)m * NCTX + i] = x[i];
}

// ---------------------------------------------------------------- gradients + Adam step
__global__ __launch_bounds__(256) void grad_adam(const float* __restrict__ Kbuf,
                                                 const float* __restrict__ alpha,
                                                 float* __restrict__ par,
                                                 float* __restrict__ am,
                                                 float* __restrict__ av,
                                                 int N, float tstep) {
  __shared__ float red[256];
  int m = blockIdx.x;
  int tid = threadIdx.x;
  const float* L = Kbuf + (size_t)m * NCTX * NCTX;
  const float* a = alpha + (size_t)m * NCTX;
  float s_alpha = 0.f, s_quad = 0.f, s_trinv = 0.f, s_logdet = 0.f;
  for (int i = tid; i < N; i += 256) {
    float ai = a[i];
    float Lii = L[(size_t)i * N + i];
    s_alpha += ai;
    s_quad += ai * ai;
    s_trinv += 1.f / (Lii * Lii);
    s_logdet += logf(fmaxf(Lii, 1e-20f));
  }
  s_alpha = block_reduce256(s_alpha, red, tid);
  s_quad = block_reduce256(s_quad, red, tid);
  s_trinv = block_reduce256(s_trinv, red, tid);
  s_logdet = block_reduce256(s_logdet, red, tid);
  if (tid == 0) {
    float inv_n = 1.f / (float)N;
    float sig0 = 1.f / (1.f + expf(-par[m * 4 + 0]));
    float sig1 = 1.f / (1.f + expf(-par[m * 4 + 1]));
    float sig2 = 1.f / (1.f + expf(-par[m * 4 + 2]));
    float g[4];
    g[0] = 0.5f * (s_trinv - s_quad) * inv_n * sig0;
    g[1] = 0.5f * (s_quad - s_trinv - 2.f * s_logdet * inv_n) * inv_n * sig1;
    g[2] = 0.5f * (s_trinv - s_quad) * inv_n * sig2;
    g[3] = -s_alpha * inv_n;
    float bc1 = 1.f - powf(BETA1, tstep);
    float bc2 = 1.f - powf(BETA2, tstep);
    for (int p = 0; p < 4; ++p) {
      int ix = m * 4 + p;
      am[ix] = BETA1 * am[ix] + (1.f - BETA1) * g[p];
      av[ix] = BETA2 * av[ix] + (1.f - BETA2) * g[p] * g[p];
      par[ix] -= LRATE * (am[ix] / bc1) / (sqrtf(av[ix] / bc2) + EPSADAM);
    }
  }
}

// ---------------------------------------------------------------- mean = const + B^T alpha
__global__ __launch_bounds__(256) void mean_kernel(const float* __restrict__ B,
                                                   const float* __restrict__ alpha,
                                                   const float* __restrict__ par,
                                                   float* __restrict__ out) {
  int m = blockIdx.x;
  int t = blockIdx.y * 256 + threadIdx.x;
  const float* Bm = B + (size_t)m * NCTX * NTGT;
  const float* a = alpha + (size_t)m * NCTX;
  float s = par[m * 4 + 3];
  for (int n = 0; n < NCTX; ++n) s += Bm[(size_t)n * NTGT + t] * a[n];
  out[(size_t)m * NTGT + t] = s;
}

// ---------------------------------------------------------------- blocked TRSM: B <- L^{-1} B
// row-panel + diagonal staged to LDS via TDM; GEMM updates via V_WMMA_F32_16X16X4_F32
__global__ __launch_bounds__(256) void trsm_lower(const float* __restrict__ Kbuf,
                                                  float* __restrict__ Bbuf) {
  __shared__ float diag[256];
  __shared__ float rowpan[16 * NTGT];  // 16x512 row panel
  int m = blockIdx.x;
  const float* L = Kbuf + (size_t)m * NCTX * NCTX;
  float* B = Bbuf + (size_t)m * NCTX * NTGT;
  int tid = threadIdx.x, lane = tid & 31, wave = tid >> 5;
  const int nb = NCTX / NB;   // 64
  const int ntj = NTGT / NB;  // 32
  for (int k = 0; k < nb; ++k) {
    // ---- stage L_kk and raw B block-row into LDS
#if HAVE_TDM
    if (tid == 0) {
      tdm_load_2d(&L[(size_t)(k * 16) * NCTX + k * 16], lds_addr32(diag),
                  16u, 16u, (unsigned)NCTX);
      tdm_load_2d(&B[(size_t)(k * 16) * NTGT], lds_addr32(rowpan),
                  16u, (unsigned)NTGT, (unsigned)NTGT);
      __builtin_amdgcn_s_wait_tensorcnt(0);
    }
#else
    diag[tid] = L[(size_t)(k * 16 + (tid >> 4)) * NCTX + k * 16 + (tid & 15)];
    for (int e = tid; e < 16 * NTGT; e += 256)
      rowpan[e] = B[(size_t)(k * 16 + e / NTGT) * NTGT + (e % NTGT)];
#endif
    __syncthreads();
    // ---- solve the 16-row block against L_kk for all 512 columns (LDS source)
    for (int c = tid; c < NTGT; c += 256) {
      float x[16];
#pragma unroll
      for (int r = 0; r < 16; ++r) x[r] = rowpan[r * NTGT + c];
#pragma unroll
      for (int r = 0; r < 16; ++r) {
        float s = x[r];
        for (int p = 0; p < r; ++p) s -= diag[r * 16 + p] * x[p];
        x[r] = s / diag[r * 16 + r];
      }
#pragma unroll
      for (int r = 0; r < 16; ++r) {
        rowpan[r * NTGT + c] = x[r];
        B[(size_t)(k * 16 + r) * NTGT + c] = x[r];
      }
    }
    __syncthreads();
    // ---- update all tiles below: C_ij -= L_ik * Vk_j
    int ntiles = (nb - 1 - k) * ntj;
    for (int idx = wave; idx < ntiles; idx += 8) {
      int i = k + 1 + idx / ntj;
      int j = idx % ntj;
      int cl = lane & 15, ch = lane >> 4;
      float* Cp = &B[(size_t)(i * 16) * NTGT + j * 16];
      const float* Lp = &L[(size_t)(i * 16) * NCTX + k * 16];
      __builtin_prefetch((const void*)Lp, 0, 1);
      v8f c;
#pragma unroll
      for (int v = 0; v < 8; ++v) c[v] = Cp[(size_t)(v + 8 * ch) * NTGT + cl];
#pragma unroll
      for (int kk = 0; kk < 16; kk += 4) {
        int k0 = kk + (ch ? 2 : 0);
        v2f a, b;
        a.x = -Lp[(size_t)cl * NCTX + k0];
        a.y = -Lp[(size_t)cl * NCTX + k0 + 1];
        b.x = rowpan[k0 * NTGT + j * 16 + cl];
        b.y = rowpan[(k0 + 1) * NTGT + j * 16 + cl];
        c = __builtin_amdgcn_wmma_f32_16x16x4_f32(false, a, false, b,
                                                  (short)0, c, false, false);
      }
#pragma unroll
      for (int v = 0; v < 8; ++v) Cp[(size_t)(v + 8 * ch) * NTGT + cl] = c[v];
    }
    // make GEMM-update stores visible before next iteration's TDM re-read
    asm volatile("s_wait_storecnt 0x0" ::: "memory");
    __syncthreads();
  }
}

// ---------------------------------------------------------------- var = os + noise - colsum(V*V)
__global__ __launch_bounds__(256) void var_kernel(const float* __restrict__ V,
                                                  const float* __restrict__ par,
                                                  float* __restrict__ out) {
  int m = blockIdx.x;
  int t = blockIdx.y * 256 + threadIdx.x;
  const float* Vm = V + (size_t)m * NCTX * NTGT;
  float os = softplusf(par[m * 4 + 1]);
  float nz = softplusf(par[m * 4 + 2]) + NOISELB;
  float s = 0.f;
  for (int n = 0; n < NCTX; ++n) {
    float v = Vm[(size_t)n * NTGT + t];
    s += v * v;
  }
  out[(size_t)M_HEADS * NTGT + (size_t)m * NTGT + t] = os - s + nz;
}

// ---------------------------------------------------------------- host orchestration
extern "C" void kernel_launch(void* const* d_in, const int* in_sizes, int n_in,
                              void* d_out, int out_size, void* d_ws, size_t ws_size,
                              hipStream_t stream) {
  const float* xc = (const float*)d_in[0];  // M*NC*DX
  const float* yc = (const float*)d_in[1];  // M*NC*1
  const float* xt = (const float*)d_in[2];  // M*NT*DX
  float* out = (float*)d_out;               // [2, M, NT]
  float* ws = (float*)d_ws;

  float* par = ws;        // 32
  float* am = ws + 32;    // 32
  float* av = ws + 64;    // 32
  float* Kbuf = ws + 256;                                  // M*NC*NC
  float* alpha = Kbuf + (size_t)M_HEADS * NCTX * NCTX;     // M*NC
  float* Bbuf = alpha + (size_t)M_HEADS * NCTX;            // M*NC*NT

  init_params<<<1, 128, 0, stream>>>(ws);

  float tstep = 0.f;
  for (int chunk = 0; chunk < 2; ++chunk) {
    int N = (chunk == 0) ? CHUNK0 : NCTX;
    dim3 gk(N / 16, N / 16, M_HEADS);
    size_t lds = (size_t)(N / NB) * 256 * sizeof(float);
    for (int s = 0; s < NSTEPS; ++s) {
      build_K<<<gk, 256, 0, stream>>>(xc, Kbuf, par, N);
      chol_inplace<<<M_HEADS, 256, lds, stream>>>(Kbuf, N);
      solve_alpha<<<M_HEADS, 256, 0, stream>>>(Kbuf, yc, par, alpha, N);
      tstep += 1.f;
      grad_adam<<<M_HEADS, 256, 0, stream>>>(Kbuf, alpha, par, am, av, N, tstep);
    }
  }

  // final inference
  {
    dim3 gk(NCTX / 16, NCTX / 16, M_HEADS);
    size_t lds = (size_t)(NCTX / NB) * 256 * sizeof(float);
    build_K<<<gk, 256, 0, stream>>>(xc, Kbuf, par, NCTX);
    chol_inplace<<<M_HEADS, 256, lds, stream>>>(Kbuf, NCTX);
    solve_alpha<<<M_HEADS, 256, 0, stream>>>(Kbuf, yc, par, alpha, NCTX);
    build_Kct<<<dim3(NTGT / 16, NCTX / 16, M_HEADS), 256, 0, stream>>>(xc, xt, Bbuf, par);
    mean_kernel<<<dim3(M_HEADS, NTGT / 256), 256, 0, stream>>>(Bbuf, alpha, par, out);
    trsm_lower<<<M_HEADS, 256, 0, stream>>>(Kbuf, Bbuf);
    var_kernel<<<dim3(M_HEADS, NTGT / 256), 256, 0, stream>>>(Bbuf, par, out);
  }
}